// MaskedTopk_31293131718893
// MI455X (gfx1250) — compile-verified
//
#include <hip/hip_runtime.h>
#include <stdint.h>

// MaskedTopk for MI455X (gfx1250).
// corr: [4,64,64,64,64] fp32 (256MB, read once -> ~11us HBM floor @23.3TB/s)
// Strategy: 1 wave32 per (b, query-pixel) row of 4096 values.
//   - Stage row HBM -> LDS with gfx1250 async copy (GLOBAL_LOAD_ASYNC_TO_LDS_B128,
//     ASYNCcnt + s_wait_asynccnt): the CDNA5-specific data-movement path.
//   - Dual (fg/bg) 256-bin radix-select on order-preserving uint keys, with
//     candidate compaction so only ~2 full LDS scans are needed per row.
//   - Wave32 bitonic sort (shfl_xor) of the 32 winners, descending.
// No __syncthreads: waves own disjoint LDS regions; same-wave LDS is processed
// in-order by hardware (ISA 7.3), and wsync() pins the compiler's issue order.

#define WAVES 2         // waves (rows) per block; 64-thread blocks
#define CAP   512       // candidate buffer entries per list
#define K     32

typedef int v4i __attribute__((ext_vector_type(4)));
typedef __attribute__((address_space(1))) v4i GV4;   // clang prints AS(1) as "__device__"
typedef __attribute__((address_space(3))) v4i LV4;   // clang prints AS(3) as "__shared__"

#ifndef __has_builtin
#define __has_builtin(x) 0
#endif
#if __has_builtin(__builtin_amdgcn_global_load_async_to_lds_b128)
#define ASYNC_LDS 1
#else
#define ASYNC_LDS 0
#endif

// Zero-instruction wave-level ordering fence: stops the compiler from moving
// LDS ops across cross-lane handoff points (hardware keeps same-wave LDS in order).
__device__ __forceinline__ void wsync() {
  asm volatile("" ::: "memory");
#if __has_builtin(__builtin_amdgcn_wave_barrier)
  __builtin_amdgcn_wave_barrier();
#endif
}

__device__ __forceinline__ unsigned ordKey(float x) {
  unsigned u = __float_as_uint(x);
  return u ^ (((unsigned)((int)u >> 31)) | 0x80000000u);
}
__device__ __forceinline__ float invKey(unsigned k) {
  return __uint_as_float((k & 0x80000000u) ? (k ^ 0x80000000u) : ~k);
}

// Wave-level: given a 256-bin histogram in (this wave's) LDS, find the largest
// digit d such that count(digit >= d) >= remaining. *countAbove = count(digit > d).
__device__ __forceinline__ unsigned digitFind(const unsigned* hist, unsigned remaining,
                                              int lane, unsigned* countAbove) {
  unsigned s = 0;
#pragma unroll
  for (int j = 0; j < 8; ++j) s += hist[lane * 8 + j];
  unsigned S = s;  // suffix-inclusive sum over lane groups (digits >= 8*lane)
#pragma unroll
  for (int o = 1; o < 32; o <<= 1) {
    unsigned t = (unsigned)__shfl_down((int)S, o, 32);
    if (lane + o < 32) S += t;
  }
  unsigned bal = (unsigned)__ballot(S >= remaining);  // prefix of lanes (S non-increasing)
  int g = 31 - __clz((int)bal);                       // group holding the threshold digit
  unsigned Sg = (unsigned)__shfl((int)S, g, 32);
  unsigned sg = (unsigned)__shfl((int)s, g, 32);
  unsigned cum = Sg - sg;  // count of digits in groups above g
  unsigned d = (unsigned)(g * 8 + 7);
#pragma unroll
  for (int j = 7; j >= 0; --j) {               // wave-uniform broadcast reads
    unsigned c = hist[g * 8 + j];
    if (cum + c >= remaining) { d = (unsigned)(g * 8 + j); break; }
    cum += c;
  }
  *countAbove = cum;
  return d;
}

__global__ void __launch_bounds__(256)
mask_kernel(const float* __restrict__ refMask, unsigned* __restrict__ maskWords) {
  int w = blockIdx.x * blockDim.x + threadIdx.x;  // 0..511 (4 batches * 128 words)
  if (w >= 512) return;
  int b = w >> 7;
  int word = w & 127;
  const float* m = refMask + (size_t)b * 65536;  // [256,256]
  unsigned bits = 0;
  for (int j = 0; j < 32; ++j) {
    int r = word * 32 + j;
    int ry = r >> 6, rx = r & 63;
    // scale=4, align_corners=False: y = 4*ry+1.5 -> y0=4ry+1, wy=0.5 (exact)
    const float* q = m + (4 * ry + 1) * 256 + (4 * rx + 1);
    float top = 0.5f * q[0]   + 0.5f * q[1];     // mirror reference rounding order
    float bot = 0.5f * q[256] + 0.5f * q[257];
    float v = 0.5f * top + 0.5f * bot;
    if (v > 0.5f) bits |= 1u << j;
  }
  maskWords[w] = bits;
}

__global__ void __launch_bounds__(WAVES * 32)
masked_topk_kernel(const float* __restrict__ corr,
                   const unsigned* __restrict__ maskWords,
                   float* __restrict__ out) {
  __shared__ __align__(16) float s_row[WAVES][4096];   // 16KB/row staged via async copy
  __shared__ unsigned s_mask[WAVES][128];
  __shared__ unsigned s_hist[WAVES][2][256];
  __shared__ unsigned s_cand[WAVES][2][CAP];
  __shared__ unsigned s_slot[WAVES][2][K];
  __shared__ unsigned s_ctr[WAVES][8];

  const int wave = threadIdx.x >> 5;
  const int lane = threadIdx.x & 31;
  const int row  = blockIdx.x * WAVES + wave;   // 0..16383
  const int b = row >> 12;
  const int p = row & 4095;

  float*    rowBuf  = s_row[wave];
  unsigned* maskBuf = s_mask[wave];
  unsigned* ctr     = s_ctr[wave];
  unsigned (*hist)[256] = s_hist[wave];
  const float* gRow = corr + (size_t)row * 4096;

  // ---- Stage 0: stream 16KB row HBM -> LDS (gfx1250 async-to-LDS path) ----
#if ASYNC_LDS
#pragma unroll
  for (int i = 0; i < 32; ++i) {
    int c = (i << 5) + lane;  // 1024 chunks of 16B
    __builtin_amdgcn_global_load_async_to_lds_b128(
        (GV4*)(gRow + c * 4), (LV4*)(rowBuf + c * 4), 0, 0);
  }
#else
  for (int i = 0; i < 32; ++i) {
    int c = (i << 5) + lane;
    ((float4*)rowBuf)[c] = ((const float4*)gRow)[c];
  }
#endif
  // overlap: mask bits + counters while the async copy is in flight
#pragma unroll
  for (int i = 0; i < 4; ++i)
    maskBuf[(i << 5) + lane] = maskWords[b * 128 + (i << 5) + lane];
  if (lane < 8) ctr[lane] = 0;
#if ASYNC_LDS
#if __has_builtin(__builtin_amdgcn_s_wait_asynccnt)
  __builtin_amdgcn_s_wait_asynccnt(0);
#else
  asm volatile("s_wait_asynccnt 0" ::: "memory");
#endif
#endif
  wsync();   // row/mask/ctr now visible to all lanes of this wave

  // ---- Phase 1: top-byte histograms for both lists (every element counts,
  //      masked-out entries contribute as +/-0 exactly like the reference) ----
#pragma unroll
  for (int j = 0; j < 8; ++j) { hist[0][lane * 8 + j] = 0; hist[1][lane * 8 + j] = 0; }
  wsync();
  for (int i = 0; i < 128; ++i) {
    float f = rowBuf[(i << 5) + lane];
    unsigned mw = maskBuf[i];                      // broadcast read; bit index == lane
    float ind = (float)((mw >> lane) & 1u);
    unsigned kb = ordKey(f * (1.0f - ind));        // bg
    unsigned kf = ordKey(f * ind);                 // fg
    atomicAdd(&hist[0][kb >> 24], 1u);
    atomicAdd(&hist[1][kf >> 24], 1u);
  }
  wsync();
  unsigned pref[2], rem[2], d3[2];
#pragma unroll
  for (int L = 0; L < 2; ++L) {
    unsigned above;
    d3[L] = digitFind(hist[L], K, lane, &above);
    pref[L] = d3[L] << 24;
    rem[L] = K - above;
  }
  wsync();

  // ---- Phase 2: winners above the threshold bin go straight to slots;
  //      threshold-bin members are compacted into a small candidate list ----
  for (int i = 0; i < 128; ++i) {
    float f = rowBuf[(i << 5) + lane];
    unsigned mw = maskBuf[i];
    float ind = (float)((mw >> lane) & 1u);
    unsigned k2[2];
    k2[0] = ordKey(f * (1.0f - ind));
    k2[1] = ordKey(f * ind);
#pragma unroll
    for (int L = 0; L < 2; ++L) {
      unsigned top = k2[L] >> 24;
      if (top > d3[L]) {
        unsigned pos = atomicAdd(&ctr[L * 2 + 0], 1u);   // < 32 by construction
        s_slot[wave][L][pos] = k2[L];
      } else if (top == d3[L]) {
        unsigned pos = atomicAdd(&ctr[L * 2 + 1], 1u);
        if (pos < CAP) s_cand[wave][L][pos] = k2[L];
      }
    }
  }
  wsync();
  unsigned pmask[2] = {0xFF000000u, 0xFF000000u};
  unsigned candN[2];
  candN[0] = ctr[1]; candN[1] = ctr[3];                  // wave-uniform broadcast
  bool fullScan[2] = { candN[0] > CAP, candN[1] > CAP }; // rare heavy-tie fallback

  // ---- Phase 3: refine the 3 remaining bytes (candidates only, typically) ----
  for (int r = 2; r >= 0; --r) {
    const int sh = r * 8;
#pragma unroll
    for (int j = 0; j < 8; ++j) { hist[0][lane * 8 + j] = 0; hist[1][lane * 8 + j] = 0; }
    wsync();
#pragma unroll
    for (int L = 0; L < 2; ++L) {
      if (!fullScan[L]) {
        for (unsigned idx = lane; idx < candN[L]; idx += 32) {
          unsigned kk = s_cand[wave][L][idx];
          if ((kk & pmask[L]) == pref[L])
            atomicAdd(&hist[L][(kk >> sh) & 255u], 1u);
        }
      }
    }
    if (fullScan[0] || fullScan[1]) {
      for (int i = 0; i < 128; ++i) {
        float f = rowBuf[(i << 5) + lane];
        unsigned mw = maskBuf[i];
        float ind = (float)((mw >> lane) & 1u);
        unsigned kk0 = ordKey(f * (1.0f - ind));
        unsigned kk1 = ordKey(f * ind);
        if (fullScan[0] && ((kk0 & pmask[0]) == pref[0]))
          atomicAdd(&hist[0][(kk0 >> sh) & 255u], 1u);
        if (fullScan[1] && ((kk1 & pmask[1]) == pref[1]))
          atomicAdd(&hist[1][(kk1 >> sh) & 255u], 1u);
      }
    }
    wsync();
#pragma unroll
    for (int L = 0; L < 2; ++L) {
      unsigned above;
      unsigned d = digitFind(hist[L], rem[L], lane, &above);
      pref[L] |= d << sh;
      pmask[L] |= 255u << sh;
      rem[L] -= above;     // stays >= 1
    }
    wsync();
  }

  // ---- Phase 4: collect exact winners (> T) plus rem[L] ties (== T) ----
#pragma unroll
  for (int L = 0; L < 2; ++L) {
    unsigned T = pref[L];          // exact key of the k-th value
    unsigned gtTot = K - rem[L];
    if (!fullScan[L]) {
      for (unsigned idx = lane; idx < candN[L]; idx += 32) {
        unsigned kk = s_cand[wave][L][idx];
        if (kk > T) {
          unsigned pos = atomicAdd(&ctr[L * 2], 1u);
          s_slot[wave][L][pos] = kk;
        } else if (kk == T) {
          unsigned pos = atomicAdd(&ctr[4 + L], 1u);
          if (pos < rem[L]) s_slot[wave][L][gtTot + pos] = kk;
        }
      }
    } else {
      for (int i = 0; i < 128; ++i) {
        float f = rowBuf[(i << 5) + lane];
        unsigned mw = maskBuf[i];
        float ind = (float)((mw >> lane) & 1u);
        unsigned kk = (L == 0) ? ordKey(f * (1.0f - ind)) : ordKey(f * ind);
        if (kk > T && (kk >> 24) == d3[L]) {   // >bin winners already slotted in phase 2
          unsigned pos = atomicAdd(&ctr[L * 2], 1u);
          s_slot[wave][L][pos] = kk;
        } else if (kk == T) {
          unsigned pos = atomicAdd(&ctr[4 + L], 1u);
          if (pos < rem[L]) s_slot[wave][L][gtTot + pos] = kk;
        }
      }
    }
  }
  wsync();   // all 32 slots of both lists are final

  // ---- Phase 5: wave32 bitonic sort (descending) + write [B,2K,H,W] ----
  size_t outBase = ((size_t)b * 64) * 4096 + (size_t)p;
#pragma unroll
  for (int L = 0; L < 2; ++L) {
    unsigned v = s_slot[wave][L][lane];
#pragma unroll
    for (int kk = 2; kk <= 32; kk <<= 1) {
#pragma unroll
      for (int j = kk >> 1; j > 0; j >>= 1) {
        unsigned o = (unsigned)__shfl_xor((int)v, j, 32);
        bool keepMax = (((lane & j) == 0) == ((lane & kk) == 0));
        unsigned hi = v > o ? v : o;
        unsigned lo = v > o ? o : v;
        v = keepMax ? hi : lo;
      }
    }
    out[outBase + (size_t)(L * K + lane) * 4096] = invKey(v);  // L=0: bg ch 0..31
  }
}

extern "C" void kernel_launch(void* const* d_in, const int* in_sizes, int n_in,
                              void* d_out, int out_size, void* d_ws, size_t ws_size,
                              hipStream_t stream) {
  (void)in_sizes; (void)n_in; (void)out_size; (void)ws_size;
  const float* corr    = (const float*)d_in[0];    // [4,64,64,64,64] fp32
  const float* refMask = (const float*)d_in[1];    // [4,1,256,256] fp32
  float* out = (float*)d_out;                      // [4,64,64,64] fp32
  unsigned* maskWords = (unsigned*)d_ws;           // 4*128 u32 = 2KB mask bits

  mask_kernel<<<2, 256, 0, stream>>>(refMask, maskWords);
  masked_topk_kernel<<<16384 / WAVES, WAVES * 32, 0, stream>>>(corr, maskWords, out);
}